// ObliviousTreeForest_37804302139403
// MI455X (gfx1250) — compile-verified
//
#include <hip/hip_runtime.h>
#include <hip/hip_bf16.h>
#include <math.h>

// ---------------------------------------------------------------------------
// ObliviousTreeForest fused pipeline for MI455X (gfx1250, wave32, WMMA)
//
//  Shapes: B=2 S=2048 D=2048 T=12 depth=3 L=8 O=2048   R = B*S = 4096
//
//  K0 : pack Wc(48x2048) = [decision_weights ; gate_w^T], softplus temps
//  K0b: pack LeafP: K-paired leaf matrix, LeafP[p][n] = {leaf[2p][n], leaf[2p+1][n]}
//  K1 : partial logits = X @ Wc^T, K split into `ksplit` ranges
//       (V_WMMA_F32_16X16X4_F32, 2 M-tiles per wave, shared B loads,
//        compile-time trip count -> real unrolling/pipelining)
//  K2 : reduce K-partials; sigmoid/temps, gate softmax, leaf probs -> q(4096x96)
//  K3 : out = LayerNorm( q @ Leaf )  (WMMA f32, fused LN, NT stores)
// ---------------------------------------------------------------------------

typedef __attribute__((ext_vector_type(2))) float v2f;
typedef __attribute__((ext_vector_type(8))) float v8f;

#define RROWS 4096
#define DDIM  2048
#define NT    12
#define NDEPTH 3
#define NLOGIT 48   // 36 decision + 12 gate
#define KQ     96   // T * 8 leaves
#define ODIM  2048
#define LN_EPS 1e-5f

// workspace layout (floats)
#define WS_WC      0                      // 48*2048  = 98304
#define WS_TEMPS   98304                  // 36 (pad 64)
#define WS_LEAFP   98368                  // 96*2048  = 196608 (paired)
#define WS_PART    294976                 // ksplit * 4096*48
#define PART_STRIDE (RROWS * NLOGIT)      // 196608
#define Q_ELEMS    (RROWS * KQ)           // 393216

static __device__ __forceinline__ v8f wmma_f32_4(v2f a, v2f b, v8f c) {
  return __builtin_amdgcn_wmma_f32_16x16x4_f32(
      /*neg_a=*/false, a, /*neg_b=*/false, b,
      /*c_mod=*/(short)0, c, /*reuse_a=*/false, /*reuse_b=*/false);
}

// ------------------------------- K0: pack Wc -------------------------------
__global__ void k0_pack(const float* __restrict__ dec_w,     // (T,depth,D)
                        const float* __restrict__ gate_w,    // (D,T)
                        const float* __restrict__ ntl,       // (T,depth)
                        float* __restrict__ Wc,              // (48,2048)
                        float* __restrict__ temps) {         // (36)
  int idx = blockIdx.x * blockDim.x + threadIdx.x;
  if (idx < NT * NDEPTH) {
    float z = ntl[idx] + 0.5413f;
    temps[idx] = (z > 20.0f) ? z : log1pf(expf(z));
  }
  if (idx >= NLOGIT * DDIM) return;
  int row = idx / DDIM;
  int col = idx - row * DDIM;
  float v;
  if (row < NT * NDEPTH) v = dec_w[idx];            // rows 0..35: direct copy
  else                   v = gate_w[col * NT + (row - NT * NDEPTH)];
  Wc[idx] = v;
}

// --------------------------- K0b: pack LeafP -------------------------------
__global__ void k0b_leafpair(const float* __restrict__ leaf,   // (96,2048)
                             float* __restrict__ leafp) {      // (48,2048) pairs
  int idx = blockIdx.x * blockDim.x + threadIdx.x;   // 48*2048 threads
  if (idx >= (KQ / 2) * ODIM) return;
  int p = idx / ODIM;
  int n = idx - p * ODIM;
  v2f pr;
  pr.x = leaf[(size_t)(2 * p)     * ODIM + n];
  pr.y = leaf[(size_t)(2 * p + 1) * ODIM + n];
  *(v2f*)(leafp + 2 * (size_t)idx) = pr;
}

// ---------------- K1: partial logits = X @ Wc^T (WMMA f32) -----------------
// grid: 16*ksplit blocks x 256 threads. blockIdx: mg = b&15, h = b>>4.
// wave -> 2 M-tiles (32 rows), shared B loads; K range [h*KLEN, (h+1)*KLEN).
template <int KLEN>
__global__ void __launch_bounds__(256)
k1_logits(const float* __restrict__ x,
          const float* __restrict__ Wc,
          float* __restrict__ part) {        // ksplit * (4096*48)
  int lane = threadIdx.x & 31;
  int wv   = threadIdx.x >> 5;
  int mg   = blockIdx.x & 15;
  int h    = blockIdx.x >> 4;
  int lo   = lane & 15;
  int hi   = lane >> 4;
  int rowbase = (mg * 8 + wv) * 32;          // 2 M-tiles per wave
  int kbeg = h * KLEN;

  const float* xr0 = x + (size_t)(rowbase + lo) * DDIM + 2 * hi + kbeg;
  const float* xr1 = x + (size_t)(rowbase + 16 + lo) * DDIM + 2 * hi + kbeg;
  const float* b0r = Wc + (size_t)(0 * 16 + lo) * DDIM + 2 * hi + kbeg;
  const float* b1r = Wc + (size_t)(1 * 16 + lo) * DDIM + 2 * hi + kbeg;
  const float* b2r = Wc + (size_t)(2 * 16 + lo) * DDIM + 2 * hi + kbeg;

  v8f c00 = {0.f,0.f,0.f,0.f,0.f,0.f,0.f,0.f};
  v8f c01 = c00, c02 = c00, c10 = c00, c11 = c00, c12 = c00;

#pragma unroll 4
  for (int k0 = 0; k0 < KLEN; k0 += 4) {
    v2f a0 = __builtin_nontemporal_load((const v2f*)(xr0 + k0));  // streamed once
    v2f a1 = __builtin_nontemporal_load((const v2f*)(xr1 + k0));
    v2f b0 = *(const v2f*)(b0r + k0);
    v2f b1 = *(const v2f*)(b1r + k0);
    v2f b2 = *(const v2f*)(b2r + k0);
    c00 = wmma_f32_4(a0, b0, c00);
    c01 = wmma_f32_4(a0, b1, c01);
    c02 = wmma_f32_4(a0, b2, c02);
    c10 = wmma_f32_4(a1, b0, c10);
    c11 = wmma_f32_4(a1, b1, c11);
    c12 = wmma_f32_4(a1, b2, c12);
  }

  float* ph = part + (size_t)h * PART_STRIDE;
#pragma unroll
  for (int j = 0; j < 8; ++j) {
    int r0 = rowbase + j + 8 * hi;
    int r1 = r0 + 16;
    ph[r0 * NLOGIT +      lo] = c00[j];
    ph[r0 * NLOGIT + 16 + lo] = c01[j];
    ph[r0 * NLOGIT + 32 + lo] = c02[j];
    ph[r1 * NLOGIT +      lo] = c10[j];
    ph[r1 * NLOGIT + 16 + lo] = c11[j];
    ph[r1 * NLOGIT + 32 + lo] = c12[j];
  }
}

// ---------------- K2: reduce partials -> q = gate * leafprob ---------------
__global__ void k2_q(const float* __restrict__ part, int ksplit,
                     const float* __restrict__ dec_b,    // (T,depth)
                     const float* __restrict__ gate_b,   // (T)
                     const float* __restrict__ temps,    // (36)
                     float* __restrict__ q) {
  int r = blockIdx.x * blockDim.x + threadIdx.x;
  if (r >= RROWS) return;

  float lg[NLOGIT];
#pragma unroll
  for (int i = 0; i < NLOGIT; ++i) lg[i] = 0.f;
  for (int h = 0; h < ksplit; ++h) {
    const float* ph = part + (size_t)h * PART_STRIDE + (size_t)r * NLOGIT;
#pragma unroll
    for (int i = 0; i < NLOGIT; ++i) lg[i] += ph[i];
  }

  // gate softmax over 12 trees
  float gl[NT];
  float mx = -3.4e38f;
#pragma unroll
  for (int t = 0; t < NT; ++t) {
    gl[t] = lg[36 + t] + gate_b[t];
    mx = fmaxf(mx, gl[t]);
  }
  float sum = 0.f;
#pragma unroll
  for (int t = 0; t < NT; ++t) { gl[t] = expf(gl[t] - mx); sum += gl[t]; }
  float inv = 1.f / sum;

  float* qr = q + (size_t)r * KQ;
#pragma unroll
  for (int t = 0; t < NT; ++t) {
    float d[NDEPTH];
#pragma unroll
    for (int n = 0; n < NDEPTH; ++n) {
      int i = t * NDEPTH + n;
      float z = (lg[i] + dec_b[i]) / temps[i];
      d[n] = 1.f / (1.f + expf(-z));
    }
    float wt = gl[t] * inv;
#pragma unroll
    for (int l = 0; l < 8; ++l) {
      float p = ((l & 4) ? 1.f - d[0] : d[0]) *
                ((l & 2) ? 1.f - d[1] : d[1]) *
                ((l & 1) ? 1.f - d[2] : d[2]);
      qr[t * 8 + l] = wt * p;
    }
  }
}

// ---------------- K3: out = LN( q @ Leaf )  (WMMA f32 + fused LN) ----------
// grid: 256 blocks x 512 threads (16 waves). block -> 16-row M tile.
// wave wv -> 128-col N slice (8 C tiles kept in registers).
__global__ void __launch_bounds__(512)
k3_gemm_ln(const float* __restrict__ q,
           const float* __restrict__ leafp,    // (48,2048) K-paired
           const float* __restrict__ gamma,
           const float* __restrict__ beta,
           float* __restrict__ out) {
  __shared__ float lds_s [16 * 16];
  __shared__ float lds_ss[16 * 16];

  int lane = threadIdx.x & 31;
  int wv   = threadIdx.x >> 5;      // 0..15
  int lo   = lane & 15;
  int hi   = lane >> 4;
  int m0   = blockIdx.x * 16;
  int nbase = wv * 128;

  // full K=96 A tile in registers (24 K-steps of 4)
  const float* qrow = q + (size_t)(m0 + lo) * KQ;
  v2f a[24];
#pragma unroll
  for (int ks = 0; ks < 24; ++ks) a[ks] = *(const v2f*)(qrow + 4 * ks + 2 * hi);

  v8f c[8];
#pragma unroll
  for (int nt = 0; nt < 8; ++nt) {
    v8f acc = {0.f,0.f,0.f,0.f,0.f,0.f,0.f,0.f};
    const float* lp = leafp + 2 * (size_t)(nbase + nt * 16 + lo);
#pragma unroll
    for (int ks = 0; ks < 24; ++ks) {
      // pair row p = 2*ks + hi holds Leaf rows k = 4*ks+2*hi, +1
      v2f b = *(const v2f*)(lp + (size_t)(2 * ks + hi) * (2 * ODIM));
      acc = wmma_f32_4(a[ks], b, acc);
    }
    c[nt] = acc;
  }

  // per-row partial sums over this wave's 128 columns
  float s[8], ss[8];
#pragma unroll
  for (int j = 0; j < 8; ++j) {
    s[j] = 0.f; ss[j] = 0.f;
#pragma unroll
    for (int nt = 0; nt < 8; ++nt) {
      float v = c[nt][j];
      s[j] += v; ss[j] += v * v;
    }
  }
  // reduce across the 16 lanes of each half (rows differ between halves)
#pragma unroll
  for (int mask = 1; mask < 16; mask <<= 1) {
#pragma unroll
    for (int j = 0; j < 8; ++j) {
      s[j]  += __shfl_xor(s[j],  mask, 32);
      ss[j] += __shfl_xor(ss[j], mask, 32);
    }
  }
  if (lo == 0) {
#pragma unroll
    for (int j = 0; j < 8; ++j) {
      int row = j + 8 * hi;
      lds_s [row * 16 + wv] = s[j];
      lds_ss[row * 16 + wv] = ss[j];
    }
  }
  __syncthreads();

  float mean[8], rstd[8];
#pragma unroll
  for (int j = 0; j < 8; ++j) {
    int row = j + 8 * hi;
    float S = 0.f, SS = 0.f;
#pragma unroll
    for (int ww = 0; ww < 16; ++ww) {
      S  += lds_s [row * 16 + ww];
      SS += lds_ss[row * 16 + ww];
    }
    float m  = S * (1.f / (float)ODIM);
    float vr = SS * (1.f / (float)ODIM) - m * m;
    mean[j] = m;
    rstd[j] = rsqrtf(vr + LN_EPS);
  }

  // normalized writeback (single pass, NT stores: no reuse of `out`)
#pragma unroll
  for (int nt = 0; nt < 8; ++nt) {
    int col = nbase + nt * 16 + lo;
    float g  = gamma[col];
    float bt = beta[col];
#pragma unroll
    for (int j = 0; j < 8; ++j) {
      int row = m0 + j + 8 * hi;
      float v = (c[nt][j] - mean[j]) * rstd[j] * g + bt;
      __builtin_nontemporal_store(v, out + (size_t)row * ODIM + col);
    }
  }
}

// ------------------------------- launcher ----------------------------------
extern "C" void kernel_launch(void* const* d_in, const int* in_sizes, int n_in,
                              void* d_out, int out_size, void* d_ws, size_t ws_size,
                              hipStream_t stream) {
  const float* x      = (const float*)d_in[0];
  const float* dec_w  = (const float*)d_in[1];
  const float* dec_b  = (const float*)d_in[2];
  const float* leaf   = (const float*)d_in[3];
  const float* gate_w = (const float*)d_in[4];
  const float* gate_b = (const float*)d_in[5];
  const float* ntl    = (const float*)d_in[6];
  const float* gamma  = (const float*)d_in[7];
  const float* beta   = (const float*)d_in[8];
  float* out = (float*)d_out;

  float* ws     = (float*)d_ws;
  float* Wc     = ws + WS_WC;
  float* temps  = ws + WS_TEMPS;
  float* leafp  = ws + WS_LEAFP;
  float* part   = ws + WS_PART;

  // choose largest K-split that fits the workspace (ws_size fixed -> deterministic)
  int ksplit = 1;
  for (int cand = 4; cand >= 1; cand >>= 1) {
    size_t need = ((size_t)WS_PART + (size_t)cand * PART_STRIDE + Q_ELEMS) * sizeof(float);
    if (need <= ws_size) { ksplit = cand; break; }
  }
  float* qbuf = part + (size_t)ksplit * PART_STRIDE;

  k0_pack<<<(NLOGIT * DDIM + 255) / 256, 256, 0, stream>>>(dec_w, gate_w, ntl, Wc, temps);
  k0b_leafpair<<<((KQ / 2) * ODIM + 255) / 256, 256, 0, stream>>>(leaf, leafp);
  if (ksplit == 4) {
    k1_logits<DDIM / 4><<<16 * 4, 256, 0, stream>>>(x, Wc, part);
  } else if (ksplit == 2) {
    k1_logits<DDIM / 2><<<16 * 2, 256, 0, stream>>>(x, Wc, part);
  } else {
    k1_logits<DDIM><<<16, 256, 0, stream>>>(x, Wc, part);
  }
  k2_q<<<RROWS / 256, 256, 0, stream>>>(part, ksplit, dec_b, gate_b, temps, qbuf);
  k3_gemm_ln<<<RROWS / 16, 512, 0, stream>>>(qbuf, leafp, gamma, beta, out);
}